// SparseCoder_13142599926224
// MI455X (gfx1250) — compile-verified
//
#include <hip/hip_runtime.h>
#include <hip/hip_bf16.h>
#include <stdint.h>
#include <stddef.h>

// Problem constants (match reference).
#define B_TOK 2048
#define DIN   1024
#define L_LAT 32768
#define K_GRP 32
#define H_GRP (L_LAT / K_GRP)   // 1024

// GEMM tiling.
#define BM 128
#define BN 128
#define BK 16
#define LDS_STRIDE 20            // padded row stride (floats): bank-conflict free, 8B aligned
#define NT_TILES (L_LAT / BN)    // 256 column tiles total
#define TPG (H_GRP / BN)         // 8 tiles per group

typedef __attribute__((ext_vector_type(2))) float v2f;
typedef __attribute__((ext_vector_type(8))) float v8f;
typedef __attribute__((ext_vector_type(2))) int   v2i;

#define AS1 __attribute__((address_space(1)))
#define AS3 __attribute__((address_space(3)))

#ifndef __has_builtin
#define __has_builtin(x) 0
#endif

#if __has_builtin(__builtin_amdgcn_global_load_async_to_lds_b64)
#define HAVE_ASYNC 1
#else
#define HAVE_ASYNC 0
#endif

#if HAVE_ASYNC
#if __has_builtin(__builtin_amdgcn_s_wait_asynccnt)
#define WAIT_ASYNC() __builtin_amdgcn_s_wait_asynccnt(0)
#else
#define WAIT_ASYNC() asm volatile("s_wait_asynccnt 0" ::: "memory")
#endif
#endif

// ---------------------------------------------------------------------------
// Kernel 1: preacts = x @ W_enc^T (+b_enc +gumbel in epilogue), per-(row, 128-col
// tile) max/argmax written to workspace. f32 WMMA 16x16x4.
// ---------------------------------------------------------------------------
__global__ __launch_bounds__(256) void enc_gemm_kernel(
    const float* __restrict__ x, const float* __restrict__ Wenc,
    const float* __restrict__ benc, const float* __restrict__ gumbel,
    float* __restrict__ tileMax, int* __restrict__ tileArg)
{
  __shared__ __align__(16) float As[2][BM][LDS_STRIDE];
  __shared__ __align__(16) float Bs[2][BN][LDS_STRIDE];
  __shared__ float redV[BM][4];
  __shared__ int   redI[BM][4];

  const int tid  = threadIdx.x;
  const int lane = tid & 31;
  const int wave = tid >> 5;     // 0..7
  const int wy   = wave >> 2;    // 0..1 -> 64 rows each
  const int wx   = wave & 3;     // 0..3 -> 32 cols each
  const int l16  = lane & 15;
  const int half = lane >> 4;    // 0/1: selects K pair (A) and K pair (B)
  const int row0 = blockIdx.y * BM;
  const int col0 = blockIdx.x * BN;
  const int mBase = wy * 64;
  const int nBase = wx * 32;

  v8f acc[4][2] = {};

  // Global -> LDS tile copy: 128x16 floats per array, as 1024 float2's / 256 thr.
  auto issueTiles = [&](int stage, int kt) {
#pragma unroll
    for (int p = 0; p < 4; ++p) {
      const int idx = (p << 8) + tid;
      const int rr = idx >> 3;            // 0..127
      const int cc = (idx & 7) << 1;      // 0,2,..,14
      const float* gx = x    + (size_t)(row0 + rr) * DIN + kt + cc;
      const float* gw = Wenc + (size_t)(col0 + rr) * DIN + kt + cc;
      float* lx = &As[stage][rr][cc];
      float* lw = &Bs[stage][rr][cc];
#if HAVE_ASYNC
      __builtin_amdgcn_global_load_async_to_lds_b64((AS1 v2i*)gx, (AS3 v2i*)lx, 0, 0);
      __builtin_amdgcn_global_load_async_to_lds_b64((AS1 v2i*)gw, (AS3 v2i*)lw, 0, 0);
#else
      *(float2*)lx = *(const float2*)gx;
      *(float2*)lw = *(const float2*)gw;
#endif
    }
  };

  issueTiles(0, 0);
  const int nk = DIN / BK;   // 64 K-steps
  for (int kti = 0; kti < nk; ++kti) {
    const int stage = kti & 1;
#if HAVE_ASYNC
    WAIT_ASYNC();
#endif
    __syncthreads();                                  // stage data ready, prev compute done
    if (kti + 1 < nk) issueTiles(stage ^ 1, (kti + 1) * BK);

#pragma unroll
    for (int kk = 0; kk < BK; kk += 4) {
      const int ka = kk + (half << 1);                // this lane's K pair
      v2f a[4], b[2];
#pragma unroll
      for (int i = 0; i < 4; ++i)
        a[i] = *(const v2f*)&As[stage][mBase + i * 16 + l16][ka];
#pragma unroll
      for (int j = 0; j < 2; ++j)
        b[j] = *(const v2f*)&Bs[stage][nBase + j * 16 + l16][ka];
#pragma unroll
      for (int i = 0; i < 4; ++i)
#pragma unroll
        for (int j = 0; j < 2; ++j)
          acc[i][j] = __builtin_amdgcn_wmma_f32_16x16x4_f32(
              false, a[i], false, b[j], (short)0, acc[i][j], false, false);
    }
  }

  // Epilogue: add b_enc + gumbel, per-row max/argmax over this 128-col tile.
  // C layout: element (M = v + 8*half, N = l16) in acc[i][j][v].
#pragma unroll
  for (int i = 0; i < 4; ++i) {
#pragma unroll
    for (int v = 0; v < 8; ++v) {
      const int rowL = mBase + i * 16 + (half << 3) + v;
      const int grow = row0 + rowL;
      float best = -__builtin_inff();
      int bestc = 0;
#pragma unroll
      for (int j = 0; j < 2; ++j) {
        const int gcol = col0 + nBase + j * 16 + l16;
        const float val = acc[i][j][v] + benc[gcol] + gumbel[(size_t)grow * L_LAT + gcol];
        if (val > best) { best = val; bestc = gcol; }   // j ascending -> first-index ties
      }
      // Reduce across 16 lanes of this half (masks <16 stay within the half).
#pragma unroll
      for (int m = 1; m < 16; m <<= 1) {
        const float ov = __shfl_xor(best, m, 32);
        const int   oc = __shfl_xor(bestc, m, 32);
        if (ov > best || (ov == best && oc < bestc)) { best = ov; bestc = oc; }
      }
      if (l16 == 0) { redV[rowL][wx] = best; redI[rowL][wx] = bestc; }
    }
  }
  __syncthreads();
  if (tid < BM) {
    float best = redV[tid][0];
    int bestc = redI[tid][0];
#pragma unroll
    for (int w = 1; w < 4; ++w) {
      const float ov = redV[tid][w];
      const int   oc = redI[tid][w];
      if (ov > best || (ov == best && oc < bestc)) { best = ov; bestc = oc; }
    }
    tileMax[(size_t)(row0 + tid) * NT_TILES + blockIdx.x] = best;
    tileArg[(size_t)(row0 + tid) * NT_TILES + blockIdx.x] = bestc;
  }
}

// ---------------------------------------------------------------------------
// Kernel 2: reduce 8 column tiles per (row, group) -> values, indices, winners.
// ---------------------------------------------------------------------------
__global__ __launch_bounds__(256) void group_reduce_kernel(
    const float* __restrict__ tileMax, const int* __restrict__ tileArg,
    float* __restrict__ outVal, float* __restrict__ outIdx, int* __restrict__ wsWin)
{
  const int id = blockIdx.x * 256 + threadIdx.x;      // < B*K
  const int row = id >> 5;
  const int g = id & 31;
  const float* tm = tileMax + (size_t)row * NT_TILES + g * TPG;
  const int*   ta = tileArg + (size_t)row * NT_TILES + g * TPG;
  float best = tm[0];
  int bestc = ta[0];
#pragma unroll
  for (int t = 1; t < TPG; ++t) {
    const float v = tm[t];
    if (v > best) { best = v; bestc = ta[t]; }        // ascending tiles => first-index ties
  }
  outVal[id] = best;
  outIdx[id] = (float)bestc;                          // global latent id (= in-group argmax + g*H)
  wsWin[id] = bestc;
}

// ---------------------------------------------------------------------------
// Kernel 3: decoded[row] = b_dec + sum_g W_dec[winner(row,g)]. Gather-sum.
// ---------------------------------------------------------------------------
__global__ __launch_bounds__(256) void decode_kernel(
    const float* __restrict__ Wdec, const float* __restrict__ bdec,
    const int* __restrict__ wsWin, float* __restrict__ outDec)
{
  const int row = blockIdx.x;
  const int c = threadIdx.x << 2;                     // 4 cols per thread (DIN=1024)
  const int* win = wsWin + (size_t)row * K_GRP;
  float4 a = *(const float4*)&bdec[c];
#pragma unroll 4
  for (int g = 0; g < K_GRP; ++g) {
    const size_t base = (size_t)win[g] * DIN + c;
    if (g + 1 < K_GRP)
      __builtin_prefetch(&Wdec[(size_t)win[g + 1] * DIN + c], 0, 1);  // global_prefetch_b8
    const float4 w = *(const float4*)&Wdec[base];
    a.x += w.x; a.y += w.y; a.z += w.z; a.w += w.w;
  }
  *(float4*)&outDec[(size_t)row * DIN + c] = a;
}

// ---------------------------------------------------------------------------
extern "C" void kernel_launch(void* const* d_in, const int* in_sizes, int n_in,
                              void* d_out, int out_size, void* d_ws, size_t ws_size,
                              hipStream_t stream) {
  (void)in_sizes; (void)n_in; (void)out_size; (void)ws_size;
  const float* x      = (const float*)d_in[0];
  const float* Wenc   = (const float*)d_in[1];
  const float* benc   = (const float*)d_in[2];
  const float* Wdec   = (const float*)d_in[3];
  const float* bdec   = (const float*)d_in[4];
  const float* gumbel = (const float*)d_in[5];
  // d_in[6] = k (constant 32, baked in)

  float* tileMax = (float*)d_ws;                                  // B*256 f32 (2 MB)
  int*   tileArg = (int*)(tileMax + (size_t)B_TOK * NT_TILES);    // B*256 i32 (2 MB)
  int*   wsWin   = tileArg + (size_t)B_TOK * NT_TILES;            // B*K   i32 (256 KB)

  float* outDec = (float*)d_out;                                  // [B, DIN]
  float* outVal = outDec + (size_t)B_TOK * DIN;                   // [B, K]
  float* outIdx = outVal + (size_t)B_TOK * K_GRP;                 // [B, K] (as f32)

  dim3 g1(NT_TILES, B_TOK / BM);   // (256, 16)
  enc_gemm_kernel<<<g1, 256, 0, stream>>>(x, Wenc, benc, gumbel, tileMax, tileArg);
  group_reduce_kernel<<<(B_TOK * K_GRP) / 256, 256, 0, stream>>>(tileMax, tileArg,
                                                                 outVal, outIdx, wsWin);
  decode_kernel<<<B_TOK, 256, 0, stream>>>(Wdec, bdec, wsWin, outDec);
}